// WD_GCN_21878563406449
// MI455X (gfx1250) — compile-verified
//
#include <hip/hip_runtime.h>
#include <math.h>

typedef float v2f __attribute__((ext_vector_type(2)));
typedef float v8f __attribute__((ext_vector_type(8)));

#define T_DIM 16
#define N_DIM 50000
#define F_DIM 64
#define H_DIM 64
#define M_NNZ 2000000
#define E_EDGE 1000000

// ---------------------------------------------------------------------------
// fp32 WMMA: D(16x16,f32) = A(16x4,f32) * B(4x16,f32) + C
// A layout: lane = khalf*16 + m (m=0..15), vgpr0 = K=k0+2*khalf, vgpr1 = K+1
// C/D layout: vgpr r -> row (r or r+8 for upper lane half), lane&15 -> col
// ---------------------------------------------------------------------------
__device__ __forceinline__ v8f wmma_f32(v2f a, v2f b, v8f c) {
    return __builtin_amdgcn_wmma_f32_16x16x4_f32(
        /*neg_a=*/false, a, /*neg_b=*/false, b,
        /*c_mod=*/(short)0, c, /*reuse_a=*/false, /*reuse_b=*/false);
}

__device__ __forceinline__ v8f splat8(float x) {
    v8f r;
#pragma unroll
    for (int i = 0; i < 8; ++i) r[i] = x;
    return r;
}

__device__ __forceinline__ v2f lo2(float4 v) { v2f r; r.x = v.x; r.y = v.y; return r; }
__device__ __forceinline__ v2f hi2(float4 v) { v2f r; r.x = v.z; r.y = v.w; return r; }

// ---------------------------------------------------------------------------
// Packer: rearrange a [K,64] row-major matrix into WMMA-B fragment order.
// One float4 per lane per K-pair: (frag(k0).x, frag(k0).y, frag(k0+4).x, .y)
// where frag(k0).x = src[(k0 + khalf*2)*64 + col], col = jt*16 + (lane&15).
// dst[frag*32 + lane], frag = k0p*ntilesTotal + tileBase + jt  -> consecutive
// lanes read a contiguous 512B block at consume time (global_load_b128).
// ---------------------------------------------------------------------------
__global__ void pack_bpairs(const float* __restrict__ src, float4* __restrict__ dst,
                            int Kpairs, int ntilesTotal, int tileBase, int ntilesThis) {
    const int idx = blockIdx.x * blockDim.x + threadIdx.x;
    const int total = Kpairs * ntilesThis * 32;
    if (idx >= total) return;
    const int lane  = idx & 31;
    const int rest  = idx >> 5;
    const int jt    = rest % ntilesThis;
    const int k0p   = rest / ntilesThis;
    const int khalf = lane >> 4;
    const int kkA   = k0p * 8 + khalf * 2;
    const int kkB   = kkA + 4;
    const int col   = jt * 16 + (lane & 15);
    float4 v;
    v.x = src[kkA * 64 + col];
    v.y = src[(kkA + 1) * 64 + col];
    v.z = src[kkB * 64 + col];
    v.w = src[(kkB + 1) * 64 + col];
    dst[(k0p * ntilesTotal + tileBase + jt) * 32 + lane] = v;
}

// ---------------------------------------------------------------------------
// K2: AX[t,row,:] += val * X[t,col,:]   (one wave per nnz, 2 floats per lane)
// ---------------------------------------------------------------------------
__global__ void spmm_scatter(const float* __restrict__ X,
                             const int* __restrict__ At, const int* __restrict__ Ar,
                             const int* __restrict__ Ac, const float* __restrict__ Av,
                             float* __restrict__ AX) {
    const int lane = threadIdx.x & 31;
    const long e = (long)blockIdx.x * 8 + (threadIdx.x >> 5);
    if (e >= M_NNZ) return;
    const int t = At[e], r = Ar[e], c = Ac[e];
    const float v = Av[e];
    const float* xr = X + ((size_t)t * N_DIM + c) * F_DIM;
    float*       ar = AX + ((size_t)t * N_DIM + r) * F_DIM;
    v2f x2 = *(const v2f*)(xr + lane * 2);
    atomicAdd(ar + lane * 2,     v * x2.x);
    atomicAdd(ar + lane * 2 + 1, v * x2.y);
}

// ---------------------------------------------------------------------------
// K3: Y = relu(AX @ W)   rows = T*N, K = 64, cols = 64.  One wave = 16 rows.
// ---------------------------------------------------------------------------
__global__ void gemm_relu_xw(const float* __restrict__ AX,
                             const float4* __restrict__ Wpk,
                             float* __restrict__ Y) {
    const int lane  = threadIdx.x & 31;
    const int wave  = threadIdx.x >> 5;
    const int m     = lane & 15;
    const int khalf = lane >> 4;
    const long mbase = ((long)blockIdx.x * 8 + wave) * 16;

    v8f acc[4];
#pragma unroll
    for (int j = 0; j < 4; ++j) acc[j] = splat8(0.f);

    const float* arow = AX + (size_t)(mbase + m) * 64;
    for (int k0p = 0; k0p < 8; ++k0p) {
        const int kk = k0p * 8 + khalf * 2;
        v2f a0 = *(const v2f*)(arow + kk);
        v2f a1 = *(const v2f*)(arow + kk + 4);
#pragma unroll
        for (int j = 0; j < 4; ++j) {
            float4 b4 = Wpk[(k0p * 4 + j) * 32 + lane];
            acc[j] = wmma_f32(a0, lo2(b4), acc[j]);
            acc[j] = wmma_f32(a1, hi2(b4), acc[j]);
        }
    }
#pragma unroll
    for (int j = 0; j < 4; ++j)
#pragma unroll
        for (int r = 0; r < 8; ++r) {
            const long row = mbase + r + khalf * 8;
            Y[(size_t)row * 64 + j * 16 + m] = fmaxf(acc[j][r], 0.f);
        }
}

// ---------------------------------------------------------------------------
// K4: LSTM scan over T. One wave owns 16 nodes for all timesteps.
// c-state stays in registers (C/D layout); h round-trips via LDS so it can be
// re-read as A fragments. Gate order (f, j, o, c) as in the reference concat.
// Gate weights come pre-packed in fragment order (one b128 feeds 2 WMMAs).
// ---------------------------------------------------------------------------
#define LSTRIDE 66
__global__ void lstm_scan(const float* __restrict__ Y, float* __restrict__ Z,
                          const float4* __restrict__ Wp, const float4* __restrict__ Up,
                          const float* __restrict__ bfv, const float* __restrict__ bjv,
                          const float* __restrict__ bov, const float* __restrict__ bcv,
                          const float* __restrict__ h_init, const float* __restrict__ c_init) {
    __shared__ __align__(16) float sY[4][16 * LSTRIDE];
    __shared__ __align__(16) float sH[4][16 * LSTRIDE];

    const int lane  = threadIdx.x & 31;
    const int wave  = threadIdx.x >> 5;
    const int m     = lane & 15;
    const int khalf = lane >> 4;
    const long tilebase = ((long)blockIdx.x * 4 + wave) * 16;

    const float* bg_[4] = {bfv, bjv, bov, bcv};

    // init state
    v8f cst[4];
#pragma unroll
    for (int q = 0; q < 4; ++q) cst[q] = splat8(c_init[q * 16 + m]);
    for (int idx = lane; idx < 16 * 64; idx += 32)
        sH[wave][(idx >> 6) * LSTRIDE + (idx & 63)] = h_init[idx & 63];
    __syncthreads();

    for (int t = 0; t < T_DIM; ++t) {
        // stage this wave's Y tile (clamped rows keep EXEC full later)
        for (int idx = lane; idx < 16 * 64; idx += 32) {
            long n = tilebase + (idx >> 6);
            if (n >= N_DIM) n = N_DIM - 1;
            sY[wave][(idx >> 6) * LSTRIDE + (idx & 63)] =
                Y[((size_t)t * N_DIM + n) * 64 + (idx & 63)];
        }
        __syncthreads();

        v8f g[16];
#pragma unroll
        for (int i = 0; i < 16; ++i) g[i] = splat8(bg_[i >> 2][(i & 3) * 16 + m]);

        const float* ya = &sY[wave][m * LSTRIDE];
        const float* ha = &sH[wave][m * LSTRIDE];
        for (int k0p = 0; k0p < 8; ++k0p) {
            const int kk = k0p * 8 + khalf * 2;
            v2f ay0 = *(const v2f*)(ya + kk);
            v2f ay1 = *(const v2f*)(ya + kk + 4);
            v2f ah0 = *(const v2f*)(ha + kk);
            v2f ah1 = *(const v2f*)(ha + kk + 4);
#pragma unroll
            for (int i = 0; i < 16; ++i) {
                float4 bw = Wp[(k0p * 16 + i) * 32 + lane];
                float4 bu = Up[(k0p * 16 + i) * 32 + lane];
                g[i] = wmma_f32(ay0, lo2(bw), g[i]);
                g[i] = wmma_f32(ay1, hi2(bw), g[i]);
                g[i] = wmma_f32(ah0, lo2(bu), g[i]);
                g[i] = wmma_f32(ah1, hi2(bu), g[i]);
            }
        }

        // all gates use sigmoid (as in the reference)
#pragma unroll
        for (int i = 0; i < 16; ++i)
#pragma unroll
            for (int r = 0; r < 8; ++r)
                g[i][r] = 1.f / (1.f + __expf(-g[i][r]));

        __syncthreads();  // done reading sH; about to overwrite it

        // c = j*ct + f*c ; h = o*tanh(c)
#pragma unroll
        for (int q = 0; q < 4; ++q) {
#pragma unroll
            for (int r = 0; r < 8; ++r) {
                const float cn = g[4 + q][r] * g[12 + q][r] + g[q][r] * cst[q][r];
                cst[q][r] = cn;
                const float h = g[8 + q][r] * tanhf(cn);
                const int rr = r + khalf * 8;
                sH[wave][rr * LSTRIDE + q * 16 + m] = h;
                const long n = tilebase + rr;
                if (n < N_DIM)
                    Z[((size_t)t * N_DIM + n) * 64 + q * 16 + m] = h;
            }
        }
        __syncthreads();
    }
}

// ---------------------------------------------------------------------------
// K5: out[e,:] = concat(Z[t,src], Z[t,trg]) @ U   ([E,128] @ [128,64])
// ---------------------------------------------------------------------------
#define ESTRIDE 132
__global__ void edge_gemm(const float* __restrict__ Z,
                          const int* __restrict__ Et, const int* __restrict__ Es,
                          const int* __restrict__ Eg, const float4* __restrict__ Upk,
                          float* __restrict__ out) {
    __shared__ __align__(16) float sA[4][16 * ESTRIDE];
    const int lane  = threadIdx.x & 31;
    const int wave  = threadIdx.x >> 5;
    const int m     = lane & 15;
    const int khalf = lane >> 4;
    const long ebase = ((long)blockIdx.x * 4 + wave) * 16;

    // gather: each lane copies one 64-float half-row (src or trg) of one edge
    {
        const int i   = lane >> 1;   // edge within tile
        const int sel = lane & 1;    // 0 = src half, 1 = trg half
        long e = ebase + i;
        if (e >= E_EDGE) e = E_EDGE - 1;
        const int t    = Et[e];
        const int node = sel ? Eg[e] : Es[e];
        const float4* zr  = (const float4*)(Z + ((size_t)t * N_DIM + node) * 64);
        float4*       dst = (float4*)(&sA[wave][i * ESTRIDE + sel * 64]);
#pragma unroll
        for (int jj = 0; jj < 16; ++jj) dst[jj] = zr[jj];
    }
    __syncthreads();

    v8f acc[4];
#pragma unroll
    for (int j = 0; j < 4; ++j) acc[j] = splat8(0.f);

    const float* ar = &sA[wave][m * ESTRIDE];
    for (int k0p = 0; k0p < 16; ++k0p) {
        const int kk = k0p * 8 + khalf * 2;
        v2f a0 = *(const v2f*)(ar + kk);
        v2f a1 = *(const v2f*)(ar + kk + 4);
#pragma unroll
        for (int j = 0; j < 4; ++j) {
            float4 b4 = Upk[(k0p * 4 + j) * 32 + lane];
            acc[j] = wmma_f32(a0, lo2(b4), acc[j]);
            acc[j] = wmma_f32(a1, hi2(b4), acc[j]);
        }
    }
#pragma unroll
    for (int j = 0; j < 4; ++j)
#pragma unroll
        for (int r = 0; r < 8; ++r) {
            const long e = ebase + r + khalf * 8;
            if (e < E_EDGE) out[(size_t)e * 64 + j * 16 + m] = acc[j][r];
        }
}

// ---------------------------------------------------------------------------
extern "C" void kernel_launch(void* const* d_in, const int* in_sizes, int n_in,
                              void* d_out, int out_size, void* d_ws, size_t ws_size,
                              hipStream_t stream) {
    const float* X   = (const float*)d_in[0];
    const int*   At  = (const int*)d_in[1];
    const int*   Ar  = (const int*)d_in[2];
    const int*   Ac  = (const int*)d_in[3];
    const float* Av  = (const float*)d_in[4];
    const int*   Et  = (const int*)d_in[5];
    const int*   Es  = (const int*)d_in[6];
    const int*   Eg  = (const int*)d_in[7];
    const float* W   = (const float*)d_in[8];
    const float* Wf  = (const float*)d_in[9];
    const float* Wj  = (const float*)d_in[10];
    const float* Wc  = (const float*)d_in[11];
    const float* Wo  = (const float*)d_in[12];
    const float* Uf  = (const float*)d_in[13];
    const float* Uj  = (const float*)d_in[14];
    const float* Uc  = (const float*)d_in[15];
    const float* Uo  = (const float*)d_in[16];
    const float* bfv = (const float*)d_in[17];
    const float* bjv = (const float*)d_in[18];
    const float* bcv = (const float*)d_in[19];
    const float* bov = (const float*)d_in[20];
    const float* h0  = (const float*)d_in[21];
    const float* c0  = (const float*)d_in[22];
    const float* Umat= (const float*)d_in[23];

    const size_t axElems = (size_t)T_DIM * N_DIM * F_DIM;  // 51.2M floats
    float*  AX   = (float*)d_ws;
    float*  Yb   = AX + axElems;
    float*  Zb   = AX;                      // AX is dead after K3; reuse for Z
    float*  pk   = Yb + axElems;            // packed-B region
    float4* Wpk  = (float4*)pk;             // 32 frags  (16 KB)
    float4* WgPk = Wpk + 32 * 32;           // 128 frags (64 KB)
    float4* UgPk = WgPk + 128 * 32;         // 128 frags (64 KB)
    float4* Upk  = UgPk + 128 * 32;         // 64 frags  (32 KB)

    // --- pack stationary B matrices into WMMA fragment order ---------------
    pack_bpairs<<<4, 256, 0, stream>>>(W, Wpk, 8, 4, 0, 4);
    const float* Wg[4] = {Wf, Wj, Wo, Wc};  // gate order (f, j, o, c)
    const float* Ug[4] = {Uf, Uj, Uo, Uc};
    for (int gate = 0; gate < 4; ++gate) {
        pack_bpairs<<<4, 256, 0, stream>>>(Wg[gate], WgPk, 8, 16, gate * 4, 4);
        pack_bpairs<<<4, 256, 0, stream>>>(Ug[gate], UgPk, 8, 16, gate * 4, 4);
    }
    pack_bpairs<<<8, 256, 0, stream>>>(Umat, Upk, 16, 4, 0, 4);

    hipMemsetAsync(AX, 0, axElems * sizeof(float), stream);

    // K2: 8 nnz per 256-thread block (one wave each)
    spmm_scatter<<<(M_NNZ + 7) / 8, 256, 0, stream>>>(X, At, Ar, Ac, Av, AX);

    // K3: 800000 rows / 16 per wave / 8 waves per block = 6250 blocks
    gemm_relu_xw<<<(T_DIM * N_DIM / 16) / 8, 256, 0, stream>>>(AX, Wpk, Yb);

    // K4: 3125 node-tiles, 4 waves per block
    lstm_scan<<<(3125 + 3) / 4, 128, 0, stream>>>(Yb, Zb, WgPk, UgPk,
                                                  bfv, bjv, bov, bcv, h0, c0);

    // K5: 62500 edge-tiles, 4 waves per block
    edge_gemm<<<(E_EDGE / 16) / 4, 128, 0, stream>>>(Zb, Et, Es, Eg, Upk,
                                                     (float*)d_out);
}